// LuTModule_68659347194173
// MI455X (gfx1250) — compile-verified
//
#include <hip/hip_runtime.h>
#include <hip/hip_bf16.h>

// -----------------------------------------------------------------------------
// LUT piecewise-linear interpolation, MI455X (gfx1250).
// Memory-bound: 402 MB of HBM traffic -> ~17us floor @ 23.3 TB/s.
// Strategy: build normalized LUT {y[i], y[i+1]-y[i]} in scratch (tiny kernel),
// then stream x -> out with b128 non-temporal loads/stores, LUT resident in LDS
// (loaded via the gfx1250 async global->LDS instruction, ASYNCcnt-tracked).
// -----------------------------------------------------------------------------

#define LUT_K 64   // reference K = 64 (65-entry table per channel)

typedef __attribute__((ext_vector_type(4))) float v4f;

__device__ __forceinline__ float softplus_stable(float v) {
    // jax.nn.softplus = max(v,0) + log1p(exp(-|v|))
    return fmaxf(v, 0.0f) + log1pf(expf(-fabsf(v)));
}

// ---------------------------------------------------------------------------
// Kernel 1: per (b,c) channel, softplus -> cumsum -> normalize, emit 64 pairs
// {y_i, y_{i+1}-y_i}. 48 channels total: trivially cheap.
// ---------------------------------------------------------------------------
__global__ void build_lut_kernel(const float* __restrict__ u,
                                 float2* __restrict__ lut, int nch) {
    int c = blockIdx.x * blockDim.x + threadIdx.x;
    if (c >= nch) return;
    const float* uc = u + (size_t)c * (LUT_K + 1);

    // y_i - y_0 = sum_{j=1..i} softplus(u_j);  y_K - y_0 = sum_{j=1..K}
    float total = 0.0f;
    for (int j = 1; j <= LUT_K; ++j) total += softplus_stable(uc[j]);
    float inv = 1.0f / total;

    float running = 0.0f;
    float prev = 0.0f;  // normalized y_i, starts at 0
    for (int i = 0; i < LUT_K; ++i) {
        running += softplus_stable(uc[i + 1]);
        float yi1 = running * inv;                     // normalized y_{i+1}
        lut[(size_t)c * LUT_K + i] = make_float2(prev, yi1 - prev);
        prev = yi1;
    }
}

// ---------------------------------------------------------------------------
// gfx1250 async global->LDS support (builtin signature: V2i*1, V2i*3, Ii, Ii)
// ---------------------------------------------------------------------------
#if __has_builtin(__builtin_amdgcn_global_load_async_to_lds_b64)
#define HAVE_ASYNC_LDS 1
typedef __attribute__((ext_vector_type(2))) int v2i_t;
typedef __attribute__((address_space(1))) v2i_t* as1_v2i_ptr;
typedef __attribute__((address_space(3))) v2i_t* as3_v2i_ptr;
#else
#define HAVE_ASYNC_LDS 0
#endif

__device__ __forceinline__ void wait_asynccnt0() {
#if __has_builtin(__builtin_amdgcn_s_wait_asynccnt)
    __builtin_amdgcn_s_wait_asynccnt(0);
#else
    asm volatile("s_wait_asynccnt 0" ::: "memory");
#endif
}

// ---------------------------------------------------------------------------
// Kernel 2: streaming interpolation. grid = (chunks, nch). 256 thr = 8 waves.
// Each thread: 4 x b128 NT loads, 16 lerps via LDS ds_load_b64, 4 x b128 NT
// stores. Chunk covers 256*4 float4 = 4096 elements.
// ---------------------------------------------------------------------------
#define VPT 4  // float4 vectors per thread

__global__ __launch_bounds__(256) void lut_interp_kernel(
    const float* __restrict__ x, const float2* __restrict__ lutg,
    float* __restrict__ out, int nvec /* float4s per channel */) {
    __shared__ float2 lut_sh[LUT_K];

    const int c = blockIdx.y;

    // ---- LUT -> LDS (512 bytes) via gfx1250 async global->LDS copy. ----
    if (threadIdx.x < LUT_K) {
#if HAVE_ASYNC_LDS
        __builtin_amdgcn_global_load_async_to_lds_b64(
            (as1_v2i_ptr)(lutg + (size_t)c * LUT_K + threadIdx.x),
            (as3_v2i_ptr)&lut_sh[threadIdx.x],
            /*offset=*/0, /*cpol=*/0);
#else
        lut_sh[threadIdx.x] = lutg[(size_t)c * LUT_K + threadIdx.x];
#endif
    }
#if HAVE_ASYNC_LDS
    wait_asynccnt0();
#endif
    __syncthreads();

    const v4f* __restrict__ xv = (const v4f*)x + (size_t)c * nvec;
    v4f* __restrict__ ov = (v4f*)out + (size_t)c * nvec;

    int v = blockIdx.x * (blockDim.x * VPT) + threadIdx.x;
#pragma unroll
    for (int it = 0; it < VPT; ++it, v += blockDim.x) {
        if (v < nvec) {
            v4f xin = __builtin_nontemporal_load(xv + v);  // b128, TH=NT
            v4f r;
#pragma unroll
            for (int j = 0; j < 4; ++j) {
                float s  = xin[j] * (float)LUT_K;
                float fi = floorf(s);
                fi = fminf(fmaxf(fi, 0.0f), (float)(LUT_K - 1));
                float2 t = lut_sh[(int)fi];          // ds_load_b64
                r[j] = __builtin_fmaf(s - fi, t.y, t.x);
            }
            __builtin_nontemporal_store(r, ov + v);  // b128, TH=NT
        }
    }
}

// ---------------------------------------------------------------------------
// Host launcher
// ---------------------------------------------------------------------------
extern "C" void kernel_launch(void* const* d_in, const int* in_sizes, int n_in,
                              void* d_out, int out_size, void* d_ws,
                              size_t ws_size, hipStream_t stream) {
    const float* x = (const float*)d_in[0];          // (16,3,1024,1024) f32
    const float* u = (const float*)d_in[1];          // (16,3,65) f32
    float* out = (float*)d_out;

    const int nch = in_sizes[1] / (LUT_K + 1);       // 48 channels
    const int hw  = in_sizes[0] / nch;               // 1,048,576 elems/channel
    const int nvec = hw >> 2;                        // float4s per channel

    float2* lut = (float2*)d_ws;                     // 48*64*8 = 24,576 B

    // Build normalized LUT (one thread per channel; trivial cost).
    {
        int threads = 64;
        int blocks = (nch + threads - 1) / threads;
        build_lut_kernel<<<blocks, threads, 0, stream>>>(u, lut, nch);
    }

    // Streaming interpolation.
    {
        const int threads = 256;
        const int vecsPerBlock = threads * VPT;      // 1024 float4 = 4096 elems
        dim3 grid((nvec + vecsPerBlock - 1) / vecsPerBlock, nch);
        lut_interp_kernel<<<grid, threads, 0, stream>>>(x, lut, out, nvec);
    }
}